// RnCLoss_degrade_18416819765838
// MI455X (gfx1250) — compile-verified
//
#include <hip/hip_runtime.h>
#include <hip/hip_bf16.h>

#define N 512
#define D 128

typedef __attribute__((ext_vector_type(2))) float v2f;
typedef __attribute__((ext_vector_type(8))) float v8f;

// ---------------------------------------------------------------------------
// Kernel 0: row squared-norms (one wave32 per row) + zero the scalar output.
// ---------------------------------------------------------------------------
__global__ void rnc_norms_kernel(const float* __restrict__ F,
                                 float* __restrict__ norms,
                                 float* __restrict__ out) {
    if (blockIdx.x == 0 && threadIdx.x == 0) out[0] = 0.0f;
    const int wave = (blockIdx.x * blockDim.x + threadIdx.x) >> 5;
    const int lane = threadIdx.x & 31;
    if (wave >= N) return;
    // D = 128 floats per row = 32 float4, exactly one per lane (b128 loads).
    const float4* row = (const float4*)(F + wave * D);
    float4 v = row[lane];
    float s = v.x * v.x + v.y * v.y + v.z * v.z + v.w * v.w;
    #pragma unroll
    for (int off = 16; off > 0; off >>= 1)
        s += __shfl_xor(s, off, 32);
    if (lane == 0) norms[wave] = s;
}

// ---------------------------------------------------------------------------
// Kernel 1: Gram matrix via V_WMMA_F32_16X16X4_F32, one wave per 16x16 tile.
// logits[i,j] = -sqrt(max(n_i + n_j - 2*G[i,j], 0)) / TEMPERATURE
// ---------------------------------------------------------------------------
__global__ void rnc_gram_logits_kernel(const float* __restrict__ F,
                                       const float* __restrict__ norms,
                                       float* __restrict__ logits) {
    const int bj   = blockIdx.x;        // column tile
    const int bi   = blockIdx.y;        // row tile
    const int lane = threadIdx.x;       // 0..31, one full wave32, EXEC all-1s
    const int half = lane >> 4;         // 0 or 1
    const int m    = lane & 15;

    // A (16x4 fp32, MxK): lane<16 holds K={0,1}, lane>=16 holds K={2,3} of row m.
    // B (4x16 fp32, KxN): B[k][n] = F[n][k]  -> identical per-lane addressing.
    const float* Arow = F + (bi * 16 + m) * D + half * 2;
    const float* Brow = F + (bj * 16 + m) * D + half * 2;

    v8f acc = {};
    #pragma unroll
    for (int kk = 0; kk < D; kk += 4) {
        v2f a = *(const v2f*)(Arow + kk);   // 8B-aligned global_load_b64
        v2f b = *(const v2f*)(Brow + kk);
        acc = __builtin_amdgcn_wmma_f32_16x16x4_f32(
            /*neg_a=*/false, a, /*neg_b=*/false, b,
            /*c_mod=*/(short)0, acc, /*reuse_a=*/false, /*reuse_b=*/false);
    }

    // C/D 16x16 f32 layout: VGPR r -> M = r + 8*half, N = m.
    const int j  = bj * 16 + m;
    const float nj = norms[j];
    #pragma unroll
    for (int r = 0; r < 8; ++r) {
        const int i = bi * 16 + r + 8 * half;
        float d2 = norms[i] + nj - 2.0f * acc[r];
        d2 = fmaxf(d2, 0.0f);
        logits[i * N + j] = -sqrtf(d2) * 0.5f;   // / TEMPERATURE (=2.0)
    }
}

// ---------------------------------------------------------------------------
// Kernel 2: per-row masked softmax loss. One 256-thread block per row i.
// denom[k] = sum_j e[j] * (ld[j] >= ld[k]); e[i] = 0 excludes the diagonal.
// ---------------------------------------------------------------------------
__global__ void rnc_loss_kernel(const float* __restrict__ logits,
                                const float* __restrict__ labels,
                                float* __restrict__ out) {
    __shared__ float s_ld[N];    // label diffs for row i
    __shared__ float s_l[N];     // max-shifted logits
    __shared__ float s_e[N];     // exp(shifted), 0 on diagonal
    __shared__ float s_red[256];

    const int i    = blockIdx.x;
    const int tid  = threadIdx.x;
    const int bdim = blockDim.x;

    // labels2[x] = labels[x % 256]  (concat of labels with itself)
    const float li0 = labels[(i & 255) * 2 + 0];
    const float li1 = labels[(i & 255) * 2 + 1];

    float lmax = -INFINITY;
    for (int j = tid; j < N; j += bdim) {
        const float lj0 = labels[(j & 255) * 2 + 0];
        const float lj1 = labels[(j & 255) * 2 + 1];
        s_ld[j] = fabsf(li0 - lj0) + fabsf(li1 - lj1);
        const float lg = logits[i * N + j];
        s_l[j] = lg;
        if (j != i) lmax = fmaxf(lmax, lg);
    }
    s_red[tid] = lmax;
    __syncthreads();
    for (int off = bdim >> 1; off > 0; off >>= 1) {
        if (tid < off) s_red[tid] = fmaxf(s_red[tid], s_red[tid + off]);
        __syncthreads();
    }
    const float mrow = s_red[0];

    for (int j = tid; j < N; j += bdim) {
        const float sl = s_l[j] - mrow;
        s_l[j] = sl;
        s_e[j] = (j == i) ? 0.0f : expf(sl);
    }
    __syncthreads();

    float loss = 0.0f;
    for (int k = tid; k < N; k += bdim) {
        if (k == i) continue;
        const float ldk = s_ld[k];
        float denom = 0.0f;
        #pragma unroll 4
        for (int j = 0; j < N; ++j) {
            // uniform-address LDS broadcast reads; per-lane threshold compare
            denom += (s_ld[j] >= ldk) ? s_e[j] : 0.0f;
        }
        loss += s_l[k] - logf(denom);   // denom >= e[k] > 0, log is safe
    }
    __syncthreads();
    s_red[tid] = loss;
    __syncthreads();
    for (int off = bdim >> 1; off > 0; off >>= 1) {
        if (tid < off) s_red[tid] += s_red[tid + off];
        __syncthreads();
    }
    if (tid == 0)
        atomicAdd(out, -s_red[0] / (float)(N * (N - 1)));
}

// ---------------------------------------------------------------------------
extern "C" void kernel_launch(void* const* d_in, const int* in_sizes, int n_in,
                              void* d_out, int out_size, void* d_ws, size_t ws_size,
                              hipStream_t stream) {
    (void)in_sizes; (void)n_in; (void)out_size; (void)ws_size;
    const float* F      = (const float*)d_in[0];   // [512,128] f32
    const float* labels = (const float*)d_in[1];   // [256,2]  f32
    float* out    = (float*)d_out;                 // scalar f32
    float* logits = (float*)d_ws;                  // [512,512] f32 scratch
    float* norms  = logits + (size_t)N * N;        // [512]     f32 scratch

    // 512 waves of 32 -> 64 blocks of 256 threads
    rnc_norms_kernel<<<dim3((N * 32) / 256), 256, 0, stream>>>(F, norms, out);
    // one wave32 per 16x16 tile, 32x32 tiles
    rnc_gram_logits_kernel<<<dim3(N / 16, N / 16), 32, 0, stream>>>(F, norms, logits);
    // one block per row
    rnc_loss_kernel<<<N, 256, 0, stream>>>(logits, labels, out);
}